// CosineSimilarity_67817533604081
// MI455X (gfx1250) — compile-verified
//
#include <hip/hip_runtime.h>
#include <hip/hip_bf16.h>

// CDNA5 / gfx1250, wave32.
// Cosine similarity = row-normalized fp32 GEMM:
//   out[q][s] = (zq[q] . zs[s]) / (max(||zq[q]||,eps) * max(||zs[s]||,eps))
// Q=8192, S=4096, D=256. Output-store bound (134 MB @ 23.3 TB/s ~ 5.8 us),
// so we keep full fp32 precision and use V_WMMA_F32_16X16X4_F32.

typedef __attribute__((ext_vector_type(2))) float v2f;
typedef __attribute__((ext_vector_type(4))) float v4f;
typedef __attribute__((ext_vector_type(8))) float v8f;

#define DIMK 256      // feature dim (K)
#define BM   128      // block tile M
#define BN   128      // block tile N
#define KC   32       // K panel per LDS stage
#define LDP  36       // padded LDS row stride (floats): multiple of 4 (16B ds ops),
                      // 36*i mod 64 distinct for i in [0,16) -> conflict-free
#define EPS  1e-8f

// ---------------------------------------------------------------------------
// 1 / max(||row||, eps), one wave32 per row (row length = 256 floats).
// ---------------------------------------------------------------------------
__global__ __launch_bounds__(256) void rnorm_kernel(const float* __restrict__ x,
                                                    float* __restrict__ r,
                                                    int rows) {
  const int wave = threadIdx.x >> 5;
  const int lane = threadIdx.x & 31;
  const int row  = blockIdx.x * 8 + wave;
  if (row >= rows) return;

  const float* p = x + (size_t)row * DIMK + lane * 8;
  v4f a = *(const v4f*)p;
  v4f b = *(const v4f*)(p + 4);
  float s = a.x * a.x + a.y * a.y + a.z * a.z + a.w * a.w
          + b.x * b.x + b.y * b.y + b.z * b.z + b.w * b.w;

  // wave32 butterfly reduction
  #pragma unroll
  for (int off = 16; off > 0; off >>= 1) s += __shfl_xor(s, off, 32);

  if (lane == 0) r[row] = 1.0f / fmaxf(sqrtf(s), EPS);
}

// ---------------------------------------------------------------------------
// Tiled fp32 WMMA GEMM with fused normalization epilogue.
//   A = z_query   [M x 256] row-major
//   B = z_support [N x 256] row-major (acts as B^T: out = A * B^T)
// Block: 128x128 tile, 256 threads = 8 waves, each wave a 32x64 sub-tile
// (2x4 grid of 16x16 WMMA accumulators).
// ---------------------------------------------------------------------------
__global__ __launch_bounds__(256) void cosine_gemm_kernel(
    const float* __restrict__ A, const float* __restrict__ B,
    const float* __restrict__ rq, const float* __restrict__ rs,
    float* __restrict__ out, int M, int N) {

  __shared__ float ldsA[BM * LDP];
  __shared__ float ldsB[BN * LDP];

  const int tid   = threadIdx.x;
  const int lane  = tid & 31;
  const int wave  = tid >> 5;
  const int lr    = lane & 15;   // lane row/col within 16
  const int half  = lane >> 4;   // 0: K pair {0,1}, 1: K pair {2,3}
  const int waveM = wave & 3;    // 4 waves along M  -> 32 rows each
  const int waveN = wave >> 2;   // 2 waves along N  -> 64 cols each

  const int mBlock = blockIdx.y * BM;
  const int nBlock = blockIdx.x * BN;

  const v8f vzero = {0.f, 0.f, 0.f, 0.f, 0.f, 0.f, 0.f, 0.f};
  v8f acc[2][4];
  #pragma unroll
  for (int i = 0; i < 2; ++i)
    #pragma unroll
    for (int j = 0; j < 4; ++j) acc[i][j] = vzero;

  for (int kb = 0; kb < DIMK; kb += KC) {
    __syncthreads();
    // Stage 128x32 panels of A and B into LDS. 1024 float4 per panel,
    // 256 threads -> 4 float4 each; 8 consecutive threads read 128B of
    // one global row (coalesced), ds_store_b128 16B-aligned (LDP % 4 == 0).
    #pragma unroll
    for (int i = 0; i < 4; ++i) {
      const int idx = tid + i * 256;  // 0..1023
      const int row = idx >> 3;
      const int seg = idx & 7;
      v4f va = *(const v4f*)(A + (size_t)(mBlock + row) * DIMK + kb + seg * 4);
      *(v4f*)&ldsA[row * LDP + seg * 4] = va;
      v4f vb = *(const v4f*)(B + (size_t)(nBlock + row) * DIMK + kb + seg * 4);
      *(v4f*)&ldsB[row * LDP + seg * 4] = vb;
    }
    // Hint-prefetch the next K panel while we compute on this one.
    if (kb + KC < DIMK) {
      __builtin_prefetch(A + (size_t)(mBlock + (tid >> 1)) * DIMK + kb + KC, 0, 3);
      __builtin_prefetch(B + (size_t)(nBlock + (tid >> 1)) * DIMK + kb + KC, 0, 3);
    }
    __syncthreads();

    // 8 K-steps of 4 -> 64 v_wmma_f32_16x16x4_f32 per wave per panel.
    #pragma unroll
    for (int ks = 0; ks < KC; ks += 4) {
      // fp32 A fragment (16x4): lanes 0-15 hold K={ks,ks+1}, lanes 16-31 K={ks+2,ks+3}
      v2f af[2];
      #pragma unroll
      for (int i = 0; i < 2; ++i)
        af[i] = *(const v2f*)&ldsA[(waveM * 32 + i * 16 + lr) * LDP + ks + 2 * half];
      // fp32 B fragment (4x16), mirrored layout over N
      v2f bf[4];
      #pragma unroll
      for (int j = 0; j < 4; ++j)
        bf[j] = *(const v2f*)&ldsB[(waveN * 64 + j * 16 + lr) * LDP + ks + 2 * half];

      #pragma unroll
      for (int i = 0; i < 2; ++i)
        #pragma unroll
        for (int j = 0; j < 4; ++j)
          acc[i][j] = __builtin_amdgcn_wmma_f32_16x16x4_f32(
              /*neg_a=*/false, af[i], /*neg_b=*/false, bf[j],
              /*c_mod=*/(short)0, acc[i][j],
              /*reuse_a=*/false, /*reuse_b=*/false);
    }
  }

  // Epilogue: D layout (16x16 f32): VGPR j, lanes 0-15 -> M=j, N=lane;
  // lanes 16-31 -> M=8+j, N=lane-16. Fuse the 1/(||q||*||s||) outer scale.
  #pragma unroll
  for (int j = 0; j < 4; ++j) {
    const int n = nBlock + waveN * 64 + j * 16 + lr;
    const float rsv = rs[n];
    #pragma unroll
    for (int i = 0; i < 2; ++i) {
      const int mbase = mBlock + waveM * 32 + i * 16 + half * 8;
      #pragma unroll
      for (int jj = 0; jj < 8; ++jj) {
        const int m = mbase + jj;
        out[(size_t)m * N + n] = acc[i][j][jj] * rq[m] * rsv;
      }
    }
  }
}

extern "C" void kernel_launch(void* const* d_in, const int* in_sizes, int n_in,
                              void* d_out, int out_size, void* d_ws, size_t ws_size,
                              hipStream_t stream) {
  const float* zq = (const float*)d_in[0];  // [Q, 256]
  const float* zs = (const float*)d_in[1];  // [S, 256]
  float* out = (float*)d_out;               // [Q, S]

  const int Q = in_sizes[0] / DIMK;  // 8192
  const int S = in_sizes[1] / DIMK;  // 4096

  float* rq = (float*)d_ws;          // Q floats
  float* rs = rq + Q;                // S floats

  rnorm_kernel<<<(Q + 7) / 8, 256, 0, stream>>>(zq, rq, Q);
  rnorm_kernel<<<(S + 7) / 8, 256, 0, stream>>>(zs, rs, S);

  dim3 grid(S / BN, Q / BM);         // (32, 64) for the reference shape
  cosine_gemm_kernel<<<grid, 256, 0, stream>>>(zq, zs, rq, rs, out, Q, S);
}